// ParamModel_3204045603582
// MI455X (gfx1250) — compile-verified
//
#include <hip/hip_runtime.h>
#include <hip/hip_bf16.h>

// ---------------------------------------------------------------------------
// Problem constants (from reference): H=64, 2H=128, INNER=16, VOCAB=256,
// B=256, L=2048, LR=0.05, EPS=1e-5, inner steps = len(arange(0,2045,2)) = 1023
// ---------------------------------------------------------------------------

typedef float v2f __attribute__((ext_vector_type(2)));
typedef float v8f __attribute__((ext_vector_type(8)));

// f32 WMMA: D(16x16) = A(16x4) * B(4x16) + C ; full fp32 (matches reference).
#define WMMA_F32(A, B, C) \
  __builtin_amdgcn_wmma_f32_16x16x4_f32(false, (A), false, (B), (short)0, (C), false, false)

static __device__ __forceinline__ v2f ld2g(const float* p) {
  return *(const v2f*)p;  // 8B-aligned by construction -> global_load_b64
}

// ---------------------------------------------------------------------------
// Phase A: hs = layernorm(h + MLP(h)) for all B*L tokens.
// One wave per 16-token tile; 4 waves (128 thr) per block; 8192 blocks.
// GEMM1 (16x128x64) and GEMM2 (16x64x128) via v_wmma_f32_16x16x4_f32.
// A-operand layout: lane = m + 16g holds A[m][4c+2g+r] in reg r.
// B-operand layout: lane = n + 16g holds B[2g+r][n]   in reg r.
// C/D layout:       lane = n + 16g, reg r  -> (m = r+8g, n).
// ---------------------------------------------------------------------------
__global__ __launch_bounds__(128) void hs_kernel(
    const int* __restrict__ seq, const float* __restrict__ embed,
    const float* __restrict__ ff1w, const float* __restrict__ ff1b,
    const float* __restrict__ ff2w, const float* __restrict__ ff2b,
    const float* __restrict__ lng, const float* __restrict__ lnb,
    float* __restrict__ hs)
{
  __shared__ float stage_all[4][16 * 132];  // per-wave 16x128 staging, padded stride
  const int lane = threadIdx.x & 31;
  const int wid  = threadIdx.x >> 5;
  const int m = lane & 15;   // A row / B column / token row
  const int g = lane >> 4;   // lane half
  float* stage = stage_all[wid];

  __builtin_prefetch(ff1w, 0, 3);   // -> global_prefetch_b8 (weights are hot)
  __builtin_prefetch(ff2w, 0, 3);

  const int tile = blockIdx.x * 4 + wid;   // 0 .. 32767
  const int tok_base = tile * 16;

  const int tok = seq[tok_base + m];
  const float* hrow = embed + tok * 64;

  // Load A tile (16 tokens x 64) in WMMA A layout.
  v2f A[16];
#pragma unroll
  for (int c = 0; c < 16; ++c)
    A[c] = ld2g(hrow + 4 * c + 2 * g);

  // GEMM1: U = relu(H @ ff1_w^T + ff1_b); stage U (16x128) row-major in LDS.
#pragma unroll 1
  for (int nt = 0; nt < 8; ++nt) {
    v8f acc = {};
#pragma unroll
    for (int c = 0; c < 16; ++c) {
      // B[k][n] = ff1_w[16*nt+n][k] ; per-lane pair (k = 4c+2g, +1)
      v2f Bv = ld2g(ff1w + (16 * nt + m) * 64 + 4 * c + 2 * g);
      acc = WMMA_F32(A[c], Bv, acc);
    }
    const float bias = ff1b[16 * nt + m];
#pragma unroll
    for (int r = 0; r < 8; ++r)
      stage[(r + 8 * g) * 132 + 16 * nt + m] = fmaxf(acc[r] + bias, 0.0f);
  }
  __syncthreads();  // uniform; DS in-order within wave anyway

  // GEMM2: F = U @ ff2_w^T + ff2_b. A chunks re-loaded from LDS in A layout.
  v8f Facc[4];
#pragma unroll 1
  for (int nt = 0; nt < 4; ++nt) {
    v8f acc = {};
#pragma unroll
    for (int c = 0; c < 32; ++c) {
      const float* ap = stage + m * 132 + 4 * c + 2 * g;  // U[m][4c+2g ..]
      v2f Av = *(const v2f*)ap;                            // ds_load_b64
      v2f Bv = ld2g(ff2w + (16 * nt + m) * 128 + 4 * c + 2 * g);
      acc = WMMA_F32(Av, Bv, acc);
    }
    const float bias = ff2b[16 * nt + m];
#pragma unroll
    for (int r = 0; r < 8; ++r) acc[r] += bias;
    Facc[nt] = acc;
  }
  __syncthreads();

  // Overwrite stage with F (U fully consumed).
#pragma unroll
  for (int nt = 0; nt < 4; ++nt)
#pragma unroll
    for (int r = 0; r < 8; ++r)
      stage[(r + 8 * g) * 132 + 16 * nt + m] = Facc[nt][r];
  __syncthreads();

  // Residual + layernorm: lane handles row m, column half g (cols 32g..32g+31).
  float x[32];
  float s = 0.f, ss = 0.f;
  const float* fr = stage + m * 132 + 32 * g;
  const float* hr = hrow + 32 * g;
#pragma unroll
  for (int j = 0; j < 32; ++j) {
    float v = fr[j] + hr[j];
    x[j] = v; s += v; ss += v * v;
  }
  s  += __shfl_xor(s, 16, 32);   // combine the two halves of the row
  ss += __shfl_xor(ss, 16, 32);
  const float mu  = s * (1.0f / 64.0f);
  const float var = ss * (1.0f / 64.0f) - mu * mu;
  const float rs  = rsqrtf(var + 1e-5f);
  float* orow = hs + (size_t)(tok_base + m) * 64 + 32 * g;
#pragma unroll
  for (int j = 0; j < 32; ++j)
    orow[j] = (x[j] - mu) * rs * lng[32 * g + j] + lnb[32 * g + j];
}

// ---------------------------------------------------------------------------
// Phase B: per-sample sequential SGD (1023 steps). One 64-thread block per
// sample (256 blocks). Thread t owns h-index t: w1 column t + w2 row t in
// registers; the two "bad direction" reductions (z, da) go through LDS.
// kh/vh are software-pipelined one iteration ahead (loads overlap the 5-barrier
// compute phase) and prefetched ~8 steps (16 rows) ahead into the near cache; the stream
// is read-once so loads are non-temporal.
// ---------------------------------------------------------------------------
__global__ __launch_bounds__(64) void ttt_kernel(
    const float* __restrict__ hs,
    const float* __restrict__ fc1w, const float* __restrict__ fc1b,
    const float* __restrict__ fc2w, const float* __restrict__ fc2b,
    float* __restrict__ ctxs)
{
  const int b = blockIdx.x;
  const int t = threadIdx.x;            // 0..63
  __shared__ float part[16][68];        // padded: conflict-free f4 row reads
  __shared__ float a_s[16];
  __shared__ float dz_s[16];
  __shared__ float b1_s[16];

  float w1c[16], w2r[16];
#pragma unroll
  for (int i = 0; i < 16; ++i) w1c[i] = fc1w[i * 64 + t];   // w1[i][t]
#pragma unroll
  for (int i = 0; i < 16; ++i) w2r[i] = fc2w[t * 16 + i];   // w2[t][i]
  float b2t = fc2b[t];
  if (t < 16) b1_s[t] = fc1b[t];
  __syncthreads();

  const float* base = hs + (size_t)b * 2048 * 64;
  float zi = 0.f;  // z[t] (valid for t<16), needed for relu mask in backward

  // Pipeline prologue: step 0's kh/vh.
  float kh = __builtin_nontemporal_load(base + t);
  float vh = __builtin_nontemporal_load(base + 64 + t);

  for (int s = 0; s < 1023; ++s) {
    // Issue next iteration's loads now; consumed after 5 barriers of compute.
    const int nrow0 = (s < 1022) ? (2 * s + 2) : 2047;  // last iter: preload qh
    const int nrow1 = (s < 1022) ? (2 * s + 3) : 2047;
    const float khn = __builtin_nontemporal_load(base + nrow0 * 64 + t);
    const float vhn = __builtin_nontemporal_load(base + nrow1 * 64 + t);
    // Prefetch ~8 steps ahead (clamped in-range).
    const int pfrow = (2 * s + 18 < 2047) ? (2 * s + 18) : 2047;
    __builtin_prefetch(base + pfrow * 64 + t, 0, 0);

    // z[i] = b1[i] + sum_t w1[i][t]*kh[t]
#pragma unroll
    for (int i = 0; i < 16; ++i) part[i][t] = w1c[i] * kh;
    __syncthreads();
    if (t < 16) {
      float acc = b1_s[t];
      const float4* pr = (const float4*)&part[t][0];
#pragma unroll
      for (int j = 0; j < 16; ++j) {
        float4 p4 = pr[j];
        acc += p4.x + p4.y + p4.z + p4.w;
      }
      zi = acc;
      a_s[t] = fmaxf(acc, 0.0f);
    }
    __syncthreads();

    // pred[t] = b2[t] + sum_i w2[t][i]*a[i] ; dpred = 2*(pred-vh)/64
    float pred = b2t;
#pragma unroll
    for (int i = 0; i < 16; ++i) pred += w2r[i] * a_s[i];
    const float dpred = (pred - vh) * (2.0f / 64.0f);

    // da[i] = sum_t w2_old[t][i]*dpred[t]
#pragma unroll
    for (int i = 0; i < 16; ++i) part[i][t] = w2r[i] * dpred;
    __syncthreads();
    if (t < 16) {
      float da = 0.f;
      const float4* pr = (const float4*)&part[t][0];
#pragma unroll
      for (int j = 0; j < 16; ++j) {
        float4 p4 = pr[j];
        da += p4.x + p4.y + p4.z + p4.w;
      }
      const float dzv = (zi > 0.0f) ? da : 0.0f;
      dz_s[t] = dzv;
      b1_s[t] -= 0.05f * dzv;          // b1 update
    }
    __syncthreads();

    // Rank-1 parameter updates (all from pre-update values).
#pragma unroll
    for (int i = 0; i < 16; ++i) w2r[i] -= 0.05f * dpred * a_s[i];
    b2t -= 0.05f * dpred;
#pragma unroll
    for (int i = 0; i < 16; ++i) w1c[i] -= 0.05f * dz_s[i] * kh;
    __syncthreads();

    kh = khn;
    vh = vhn;
  }

  // ctx = relu(qh @ w1^T + b1) @ w2^T + b2 with final params; qh = hs[b, 2047]
  // (already pipelined into `kh` by the last loop iteration).
  {
#pragma unroll
    for (int i = 0; i < 16; ++i) part[i][t] = w1c[i] * kh;
    __syncthreads();
    if (t < 16) {
      float acc = b1_s[t];
      const float4* pr = (const float4*)&part[t][0];
#pragma unroll
      for (int j = 0; j < 16; ++j) {
        float4 p4 = pr[j];
        acc += p4.x + p4.y + p4.z + p4.w;
      }
      a_s[t] = fmaxf(acc, 0.0f);
    }
    __syncthreads();
    float pred = b2t;
#pragma unroll
    for (int i = 0; i < 16; ++i) pred += w2r[i] * a_s[i];
    ctxs[b * 64 + t] = pred;
  }
}

// ---------------------------------------------------------------------------
// Phase C: out = ctxs(256x64) @ out_w^T(64x256) + out_b, via WMMA 16x16 tiles.
// 256 tiles, one wave each; 64 blocks x 128 threads.
// ---------------------------------------------------------------------------
__global__ __launch_bounds__(128) void out_kernel(
    const float* __restrict__ ctxs, const float* __restrict__ outw,
    const float* __restrict__ outb, float* __restrict__ out)
{
  const int lane = threadIdx.x & 31;
  const int wid  = threadIdx.x >> 5;
  const int m = lane & 15;
  const int g = lane >> 4;
  const int tile = blockIdx.x * 4 + wid;   // 0..255
  const int tm = tile >> 4, tn = tile & 15;

  v2f A[16];
#pragma unroll
  for (int c = 0; c < 16; ++c)
    A[c] = ld2g(ctxs + (16 * tm + m) * 64 + 4 * c + 2 * g);

  v8f acc = {};
#pragma unroll
  for (int c = 0; c < 16; ++c) {
    v2f Bv = ld2g(outw + (16 * tn + m) * 64 + 4 * c + 2 * g);  // out_w[n][k]
    acc = WMMA_F32(A[c], Bv, acc);
  }
  const float bias = outb[16 * tn + m];
#pragma unroll
  for (int r = 0; r < 8; ++r)
    out[(size_t)(16 * tm + r + 8 * g) * 256 + 16 * tn + m] = acc[r] + bias;
}

// ---------------------------------------------------------------------------
extern "C" void kernel_launch(void* const* d_in, const int* in_sizes, int n_in,
                              void* d_out, int out_size, void* d_ws, size_t ws_size,
                              hipStream_t stream) {
  (void)in_sizes; (void)n_in; (void)out_size; (void)ws_size;

  const int*   seq   = (const int*)  d_in[0];
  const float* embed = (const float*)d_in[1];
  const float* ff1w  = (const float*)d_in[2];
  const float* ff1b  = (const float*)d_in[3];
  const float* ff2w  = (const float*)d_in[4];
  const float* ff2b  = (const float*)d_in[5];
  const float* lng   = (const float*)d_in[6];
  const float* lnb   = (const float*)d_in[7];
  const float* fc1w  = (const float*)d_in[8];
  const float* fc1b  = (const float*)d_in[9];
  const float* fc2w  = (const float*)d_in[10];
  const float* fc2b  = (const float*)d_in[11];
  const float* outw  = (const float*)d_in[12];
  const float* outb  = (const float*)d_in[13];

  // Workspace layout: hs fp32 [256*2048, 64] = 128 MiB, then ctxs [256,64].
  float* hs   = (float*)d_ws;
  float* ctxs = (float*)((char*)d_ws + (size_t)256 * 2048 * 64 * sizeof(float));

  hs_kernel <<<8192, 128, 0, stream>>>(seq, embed, ff1w, ff1b, ff2w, ff2b, lng, lnb, hs);
  ttt_kernel<<<256,   64, 0, stream>>>(hs, fc1w, fc1b, fc2w, fc2b, ctxs);
  out_kernel<<<64,   128, 0, stream>>>(ctxs, outw, outb, (float*)d_out);
}